// SubdivideMeshes_9818295239267
// MI455X (gfx1250) — compile-verified
//
#include <hip/hip_runtime.h>
#include <hip/hip_bf16.h>

// Problem constants (from the reference).
#define NN 16
#define VV 40000
#define EE 120000
#define FF 80000
#define DD 64

#define CHUNK_DW 16000           // dwords per TDM chunk (62.5 KB); V*D = 160 * CHUNK_DW
#define CHUNKS_PER_N 160

typedef float v4f  __attribute__((ext_vector_type(4)));
typedef unsigned int u32x4 __attribute__((ext_vector_type(4)));
typedef int i32x4 __attribute__((ext_vector_type(4)));
typedef int i32x8 __attribute__((ext_vector_type(8)));

// ---------------------------------------------------------------------------
// Kernel 1: TDM streaming copy of feats -> new_feats[:, 0:V, :]
// One wave per workgroup issues a tensor_load_to_lds + tensor_store_from_lds
// descriptor pair; the Tensor Data Mover does the 62.5KB chunk copy.
// ---------------------------------------------------------------------------
__global__ __launch_bounds__(32) void tdm_copy_feats(const float* __restrict__ src,
                                                     float* __restrict__ dst) {
    __shared__ unsigned int lds_buf[CHUNK_DW];

    const int w     = blockIdx.x;
    const int n     = w / CHUNKS_PER_N;
    const int chunk = w % CHUNKS_PER_N;

    const unsigned long long sa = reinterpret_cast<unsigned long long>(
        src + (long long)n * VV * DD + (long long)chunk * CHUNK_DW);
    const unsigned long long da = reinterpret_cast<unsigned long long>(
        dst + (long long)n * (long long)(VV + EE) * DD + (long long)chunk * CHUNK_DW);

    // Touch LDS so the 64KB group segment is actually allocated (volatile store,
    // not DCE-able); descriptor lds_addr=0 targets this buffer.
    if (threadIdx.x == 0) ((volatile unsigned int*)lds_buf)[0] = 0u;
    asm volatile("s_wait_dscnt 0" ::: "memory");

    // D# group 0: [1:0]=count=1 | [63:32]=lds_addr=0 | [120:64]=global_addr | [127:126]=type=2
    u32x4 g0_load;
    g0_load[0] = 1u;
    g0_load[1] = 0u;
    g0_load[2] = (unsigned)(sa & 0xFFFFFFFFull);
    g0_load[3] = (unsigned)((sa >> 32) & 0x01FFFFFFull) | 0x80000000u;

    u32x4 g0_store = g0_load;
    g0_store[2] = (unsigned)(da & 0xFFFFFFFFull);
    g0_store[3] = (unsigned)((da >> 32) & 0x01FFFFFFull) | 0x80000000u;

    // D# group 1: wg_mask=0, data_size=2 (4B), tensor_dim0=CHUNK_DW, tensor_dim1=1,
    // tile_dim0=CHUNK_DW (1 row), tile_dim1/2=0 (unused), tensor_dim0_stride=CHUNK_DW.
    i32x8 g1;
    g1[0] = (int)(2u << 16);                                   // data_size = 4 bytes
    g1[1] = (int)((CHUNK_DW & 0xFFFFu) << 16);                 // tensor_dim0[15:0]
    g1[2] = (int)(((CHUNK_DW >> 16) & 0xFFFFu) | (1u << 16));  // tensor_dim0[31:16] | tensor_dim1 lo = 1
    g1[3] = (int)((CHUNK_DW & 0xFFFFu) << 16);                 // tensor_dim1 hi = 0 | tile_dim0
    g1[4] = 0;                                                 // tile_dim1 = 0, tile_dim2 = 0
    g1[5] = (int)CHUNK_DW;                                     // tensor_dim0_stride lo32
    g1[6] = 0;
    g1[7] = 0;

    i32x4 z4 = {0, 0, 0, 0};
    i32x8 z8 = {0, 0, 0, 0, 0, 0, 0, 0};

    __builtin_amdgcn_tensor_load_to_lds(g0_load, g1, z4, z4, z8, 0);
    __builtin_amdgcn_s_wait_tensorcnt(0);                      // LDS chunk resident
    __builtin_amdgcn_tensor_store_from_lds(g0_store, g1, z4, z4, z8, 0);
    __builtin_amdgcn_s_wait_tensorcnt(0);                      // drained before wave ends
}

// ---------------------------------------------------------------------------
// Kernel 2: mid-edge feature mean -> new_feats[:, V:V+E, :]
// 16 lanes x float4 = one fully coalesced 256B row. Wave32 covers 2 rows.
// Gather loads stay RT (feats fits in 192MB L2); output stores are NT.
// ---------------------------------------------------------------------------
__global__ __launch_bounds__(256) void mid_feats_kernel(const float* __restrict__ feats,
                                                        const int* __restrict__ edges,
                                                        float* __restrict__ out_feats) {
    const int tid = threadIdx.x;
    const long long row = (long long)blockIdx.x * 16 + (tid >> 4);  // n*E + e
    const int c = tid & 15;                                         // float4 slot in row
    const int e = (int)(row % EE);
    const int n = (int)(row / EE);

    const int v0 = edges[2 * e + 0];
    const int v1 = edges[2 * e + 1];

    const v4f* a = reinterpret_cast<const v4f*>(feats + ((long long)n * VV + v0) * DD) + c;
    const v4f* b = reinterpret_cast<const v4f*>(feats + ((long long)n * VV + v1) * DD) + c;
    v4f m = (*a + *b) * 0.5f;

    v4f* o = reinterpret_cast<v4f*>(out_feats +
              ((long long)n * (VV + EE) + VV + e) * DD) + c;
    __builtin_nontemporal_store(m, o);
}

// ---------------------------------------------------------------------------
// Kernel 3: new_verts = concat(verts, mid_verts). Thread per output float.
// ---------------------------------------------------------------------------
__global__ __launch_bounds__(256) void new_verts_kernel(const float* __restrict__ verts,
                                                        const int* __restrict__ edges,
                                                        float* __restrict__ out_verts) {
    const long long i = (long long)blockIdx.x * 256 + threadIdx.x;  // < N*(V+E)*3
    const int c = (int)(i % 3);
    const long long r = i / 3;
    const int p = (int)(r % (VV + EE));
    const int n = (int)(r / (VV + EE));

    float val;
    if (p < VV) {
        val = verts[((long long)n * VV + p) * 3 + c];
    } else {
        const int e = p - VV;
        const int v0 = edges[2 * e + 0];
        const int v1 = edges[2 * e + 1];
        val = 0.5f * (verts[((long long)n * VV + v0) * 3 + c] +
                      verts[((long long)n * VV + v1) * 3 + c]);
    }
    __builtin_nontemporal_store(val, out_verts + i);
}

// ---------------------------------------------------------------------------
// Kernel 4: subdivided face connectivity (written as f32; indices < 2^24).
// new_faces = concat([f0, f1, f2, fe], axis=0), fe = faces_to_edges + V.
// ---------------------------------------------------------------------------
__global__ __launch_bounds__(256) void new_faces_kernel(const int* __restrict__ faces,
                                                        const int* __restrict__ fte,
                                                        float* __restrict__ out_faces) {
    const int f = blockIdx.x * 256 + threadIdx.x;
    if (f >= FF) return;

    const int a  = faces[3 * f + 0];
    const int b  = faces[3 * f + 1];
    const int cc = faces[3 * f + 2];
    const int e0 = fte[3 * f + 0] + VV;
    const int e1 = fte[3 * f + 1] + VV;
    const int e2 = fte[3 * f + 2] + VV;

    float* o0 = out_faces + 3 * f;                 // [a,  e2, e1]
    float* o1 = out_faces + 3 * (FF + f);          // [b,  e0, e2]
    float* o2 = out_faces + 3 * (2 * FF + f);      // [cc, e1, e0]
    float* o3 = out_faces + 3 * (3 * FF + f);      // [e0, e1, e2]
    o0[0] = (float)a;  o0[1] = (float)e2; o0[2] = (float)e1;
    o1[0] = (float)b;  o1[1] = (float)e0; o1[2] = (float)e2;
    o2[0] = (float)cc; o2[1] = (float)e1; o2[2] = (float)e0;
    o3[0] = (float)e0; o3[1] = (float)e1; o3[2] = (float)e2;
}

// ---------------------------------------------------------------------------
extern "C" void kernel_launch(void* const* d_in, const int* in_sizes, int n_in,
                              void* d_out, int out_size, void* d_ws, size_t ws_size,
                              hipStream_t stream) {
    const float* verts = (const float*)d_in[0];   // (N, V, 3)
    const float* feats = (const float*)d_in[1];   // (N, V, D)
    const int*   edges = (const int*)d_in[2];     // (E, 2)
    const int*   faces = (const int*)d_in[3];     // (F, 3)
    const int*   fte   = (const int*)d_in[4];     // (F, 3)

    float* out       = (float*)d_out;
    float* out_verts = out;                                          // N*(V+E)*3
    float* out_faces = out_verts + (long long)NN * (VV + EE) * 3;    // 4F*3
    float* out_feats = out_faces + 4LL * FF * 3;                     // N*(V+E)*D

    // 1) DMA the contiguous feats block into new_feats[:, 0:V, :] via TDM.
    tdm_copy_feats<<<NN * CHUNKS_PER_N, 32, 0, stream>>>(feats, out_feats);

    // 2) Gather-mean midpoint features (dominant traffic).
    mid_feats_kernel<<<(NN * EE) / 16, 256, 0, stream>>>(feats, edges, out_feats);

    // 3) Vertices (copy + midpoints).
    new_verts_kernel<<<((long long)NN * (VV + EE) * 3) / 256, 256, 0, stream>>>(
        verts, edges, out_verts);

    // 4) Face connectivity.
    new_faces_kernel<<<(FF + 255) / 256, 256, 0, stream>>>(faces, fte, out_faces);
}